// ProductKeyMemory_26749056319687
// MI455X (gfx1250) — compile-verified
//
#include <hip/hip_runtime.h>
#include <math.h>

// ---------------- problem constants ----------------
#define BB 4
#define SS 2048
#define DD 1024
#define MSLOTS 65536
#define SDIM 256
#define SKEY 128
#define TOPK 32
#define CBN 256
#define MP 16          // padded M for WMMA (rows 4..15 are zero)

// ---------------- ws layout (float offsets) ----------------
#define WS_XSP     0                    // [16,1024] padded x_summary
#define WS_QA      (WS_XSP + MP*DD)     // [16,128]
#define WS_QB      (WS_QA + MP*SKEY)    // [16,128]
#define WS_WV      (WS_QB + MP*SKEY)    // [16,256]
#define WS_SIMA    (WS_WV + MP*SDIM)    // [16,256]
#define WS_SIMB    (WS_SIMA + MP*CBN)   // [16,256]
#define WS_SA      (WS_SIMB + MP*CBN)   // [4,32] scores_a
#define WS_SB      (WS_SA + BB*TOPK)    // [4,32]
#define WS_IA      (WS_SB + BB*TOPK)    // [4,32] int
#define WS_IB      (WS_IA + BB*TOPK)    // [4,32] int
#define WS_P       (WS_IB + BB*TOPK)    // [4,1024] softmax probs
#define WS_CIDX    (WS_P + BB*TOPK*TOPK)// [4,1024] int
#define WS_PART    (WS_CIDX + BB*TOPK*TOPK) // [32,256] read_out partials
#define WS_RO      (WS_PART + 32*SDIM)  // [16,256] padded read_out
#define WS_RP      (WS_RO + MP*SDIM)    // [16,1024] read_projected
#define WS_GATE    (WS_RP + MP*DD)      // [4]

typedef float v2f __attribute__((ext_vector_type(2)));
typedef float v4f __attribute__((ext_vector_type(4)));
typedef float v8f __attribute__((ext_vector_type(8)));

// ================= kernel 1: padded x_summary =================
// RT loads on purpose: x is re-read by k_xaug and fits in the 192MB L2.
__global__ void k_xsummary(const float* __restrict__ x, float* __restrict__ ws) {
    int idx = blockIdx.x * blockDim.x + threadIdx.x;   // 0..16*1024-1
    int b = idx >> 10, d = idx & 1023;
    float acc = 0.0f;
    if (b < BB) {
        const float* xp = x + (size_t)b * SS * DD + d;
        for (int s = 0; s < SS; ++s) acc += xp[(size_t)s * DD];
        acc *= (1.0f / (float)SS);
    }
    ws[WS_XSP + idx] = acc;
}

// ================= WMMA helpers (f32 16x16x4) =================
// A: [16,Kdim] row-major (lda=Kdim).  C: [16,ldc] row-major.
// B row-major [Kdim, ldb]:
__device__ __forceinline__ void wmma_gemm_rowB(
    const float* __restrict__ A, int Kdim,
    const float* __restrict__ Bm, int ldb,
    const float* __restrict__ bias,
    float* __restrict__ C, int ldc, int n0)
{
    int lane = threadIdx.x & 31;
    int row = lane & 15;        // M for A-frag, N-offset for B-frag
    int h   = lane >> 4;        // K-half selector
    v8f c = {};
    for (int k0 = 0; k0 < Kdim; k0 += 4) {
        int ka = k0 + 2 * h;
        v2f a, b;
        a.x = A[row * Kdim + ka];
        a.y = A[row * Kdim + ka + 1];
        b.x = Bm[(size_t)ka * ldb + n0 + row];
        b.y = Bm[(size_t)(ka + 1) * ldb + n0 + row];
        c = __builtin_amdgcn_wmma_f32_16x16x4_f32(false, a, false, b,
                                                  (short)0, c, false, false);
    }
    int n = n0 + row;
    float bv = bias ? bias[n] : 0.0f;
    for (int r = 0; r < 8; ++r) {
        int m = h ? (r + 8) : r;
        C[m * ldc + n] = c[r] + bv;
    }
}

// B = codebook^T : B[k,n] = cb[n*SKEY + k]  (cb is [CBN, SKEY] row-major)
__device__ __forceinline__ void wmma_gemm_cbT(
    const float* __restrict__ A, int Kdim,
    const float* __restrict__ cb,
    float* __restrict__ C, int ldc, int n0)
{
    int lane = threadIdx.x & 31;
    int row = lane & 15;
    int h   = lane >> 4;
    v8f c = {};
    for (int k0 = 0; k0 < Kdim; k0 += 4) {
        int ka = k0 + 2 * h;
        v2f a, b;
        a.x = A[row * Kdim + ka];
        a.y = A[row * Kdim + ka + 1];
        b.x = cb[(n0 + row) * SKEY + ka];
        b.y = cb[(n0 + row) * SKEY + ka + 1];
        c = __builtin_amdgcn_wmma_f32_16x16x4_f32(false, a, false, b,
                                                  (short)0, c, false, false);
    }
    int n = n0 + row;
    for (int r = 0; r < 8; ++r) {
        int m = h ? (r + 8) : r;
        C[m * ldc + n] = c[r];
    }
}

// ===== kernel 2: q_a, q_b, write_value via WMMA (32 waves) =====
__global__ void k_wmma_qv(const float* __restrict__ Wa, const float* __restrict__ ba,
                          const float* __restrict__ Wb, const float* __restrict__ bb,
                          const float* __restrict__ Wv, const float* __restrict__ bv,
                          float* __restrict__ ws) {
    const float* xsp = ws + WS_XSP;
    int bid = blockIdx.x;
    if (bid < 8) {
        wmma_gemm_rowB(xsp, DD, Wa, SKEY, ba, ws + WS_QA, SKEY, bid * 16);
    } else if (bid < 16) {
        wmma_gemm_rowB(xsp, DD, Wb, SKEY, bb, ws + WS_QB, SKEY, (bid - 8) * 16);
    } else {
        wmma_gemm_rowB(xsp, DD, Wv, SDIM, bv, ws + WS_WV, SDIM, (bid - 16) * 16);
    }
}

// ===== kernel 3: sim_a, sim_b via WMMA (32 waves) =====
__global__ void k_wmma_sim(const float* __restrict__ cba,
                           const float* __restrict__ cbb,
                           float* __restrict__ ws) {
    int bid = blockIdx.x;
    if (bid < 16) {
        wmma_gemm_cbT(ws + WS_QA, SKEY, cba, ws + WS_SIMA, CBN, bid * 16);
    } else {
        wmma_gemm_cbT(ws + WS_QB, SKEY, cbb, ws + WS_SIMB, CBN, (bid - 16) * 16);
    }
}

// ===== kernel 4: write gate (4 blocks) =====
__global__ void k_gate(const float* __restrict__ Wg, const float* __restrict__ bg,
                       float* __restrict__ ws) {
    __shared__ float red[256];
    int b = blockIdx.x, tid = threadIdx.x;
    const float* xs = ws + WS_XSP + b * DD;
    float acc = 0.0f;
    for (int d = tid; d < DD; d += 256) acc += xs[d] * Wg[d];
    red[tid] = acc; __syncthreads();
    for (int off = 128; off > 0; off >>= 1) {
        if (tid < off) red[tid] += red[tid + off];
        __syncthreads();
    }
    if (tid == 0) ws[WS_GATE + b] = 1.0f / (1.0f + expf(-(red[0] + bg[0])));
}

// ===== kernel 5: top-32 per (batch, side), argmax iteration =====
__global__ void k_topk(float* __restrict__ ws) {
    __shared__ float work[CBN];
    __shared__ float sv[CBN];
    __shared__ int   si[CBN];
    int b = blockIdx.x >> 1, side = blockIdx.x & 1, tid = threadIdx.x;
    const float* sim = ws + (side ? WS_SIMB : WS_SIMA) + b * CBN;
    float* scores = ws + (side ? WS_SB : WS_SA) + b * TOPK;
    int*   idxs   = (int*)(ws + (side ? WS_IB : WS_IA)) + b * TOPK;
    work[tid] = sim[tid];
    __syncthreads();
    for (int t = 0; t < TOPK; ++t) {
        sv[tid] = work[tid]; si[tid] = tid;
        __syncthreads();
        for (int off = 128; off > 0; off >>= 1) {
            if (tid < off) {
                float ov = sv[tid + off]; int oi = si[tid + off];
                if (ov > sv[tid] || (ov == sv[tid] && oi < si[tid])) {
                    sv[tid] = ov; si[tid] = oi;
                }
            }
            __syncthreads();
        }
        if (tid == 0) {
            scores[t] = sv[0];
            idxs[t]   = si[0];
            work[si[0]] = -3.0e38f;
        }
        __syncthreads();
    }
}

// ===== kernel 6: combined scores + softmax + combined idx (4 blocks x 1024) =====
__global__ void k_softmax(float* __restrict__ ws) {
    __shared__ float red[1024];
    int b = blockIdx.x, tid = threadIdx.x;
    int i = tid >> 5, j = tid & 31;
    const float scale = 0.0883883476483184f;   // 1/sqrt(128)
    float sa = ws[WS_SA + b * TOPK + i];
    float sb = ws[WS_SB + b * TOPK + j];
    int   ia = ((int*)(ws + WS_IA))[b * TOPK + i];
    int   ib = ((int*)(ws + WS_IB))[b * TOPK + j];
    float l = (sa + sb) * scale;
    red[tid] = l; __syncthreads();
    for (int off = 512; off > 0; off >>= 1) {
        if (tid < off) red[tid] = fmaxf(red[tid], red[tid + off]);
        __syncthreads();
    }
    float m = red[0]; __syncthreads();
    float e = expf(l - m);
    red[tid] = e; __syncthreads();
    for (int off = 512; off > 0; off >>= 1) {
        if (tid < off) red[tid] += red[tid + off];
        __syncthreads();
    }
    float inv = 1.0f / red[0];
    ws[WS_P + b * 1024 + tid] = e * inv;
    ((int*)(ws + WS_CIDX))[b * 1024 + tid] = ia * CBN + ib;
}

// ===== kernel 7: read_out partials (4 batch x 8 chunks) =====
__global__ void k_readout_part(const float* __restrict__ memory, float* __restrict__ ws) {
    int b = blockIdx.x >> 3, chunk = blockIdx.x & 7, d = threadIdx.x;
    const float* P = ws + WS_P + b * 1024 + chunk * 128;
    const int* CI = (const int*)(ws + WS_CIDX) + b * 1024 + chunk * 128;
    const float* memb = memory + (size_t)b * MSLOTS * SDIM;
    float acc = 0.0f;
    for (int k = 0; k < 128; ++k) {
        acc += P[k] * memb[(size_t)CI[k] * SDIM + d];
    }
    ws[WS_PART + (size_t)blockIdx.x * SDIM + d] = acc;
}

// ===== kernel 8: reduce partials -> padded read_out (16 blocks) =====
__global__ void k_readout_red(float* __restrict__ ws) {
    int m = blockIdx.x, d = threadIdx.x;
    float acc = 0.0f;
    if (m < BB)
        for (int c = 0; c < 8; ++c) acc += ws[WS_PART + (m * 8 + c) * SDIM + d];
    ws[WS_RO + m * SDIM + d] = acc;
}

// ===== kernel 9: read_projected = ro @ Wo + bo via WMMA (64 waves) =====
__global__ void k_wmma_rp(const float* __restrict__ Wo, const float* __restrict__ bo,
                          float* __restrict__ ws) {
    wmma_gemm_rowB(ws + WS_RO, SDIM, Wo, DD, bo, ws + WS_RP, DD, blockIdx.x * 16);
}

// ===== kernel 10: x_augmented (64 MB stream, NT policy) =====
__global__ void k_xaug(const float* __restrict__ x, const float* __restrict__ ws,
                       float* __restrict__ out) {
    size_t idx = (size_t)blockIdx.x * blockDim.x + threadIdx.x;   // < 2097152
    int b = (int)(idx >> 19);                 // 2048*1024/4 per batch
    int d4 = (int)(idx & 255);                // 1024/4 per row
    // x: second & last use -> NT load (may still hit the L2-resident copy)
    v4f xv = __builtin_nontemporal_load((const v4f*)x + idx);
    v4f rp = ((const v4f*)(ws + WS_RP + b * DD))[d4];   // tiny, keep cached
    v4f o = xv + rp;
    __builtin_nontemporal_store(o, (v4f*)out + idx);    // streamed once
}

// ===== kernel 11: memory_new (512 MB stream, NT policy) =====
__global__ void k_memupd(const float* __restrict__ memory, const float* __restrict__ ws,
                         float* __restrict__ out) {
    size_t idx = (size_t)blockIdx.x * blockDim.x + threadIdx.x;   // < 16777216
    int b = (int)(idx >> 22);                 // 65536*256/4 per batch
    int d4 = (int)(idx & 63);                 // 256/4 per slot
    float s = 0.1f * ws[WS_GATE + b];
    v4f wv = ((const v4f*)(ws + WS_WV + b * SDIM))[d4]; // 4KB, keep cached
    v4f mv = __builtin_nontemporal_load((const v4f*)memory + idx);  // touched once
    v4f o = mv + s * wv;
    __builtin_nontemporal_store(o, (v4f*)out + idx);    // touched once
}

// ================= launch =================
extern "C" void kernel_launch(void* const* d_in, const int* in_sizes, int n_in,
                              void* d_out, int out_size, void* d_ws, size_t ws_size,
                              hipStream_t stream) {
    const float* x      = (const float*)d_in[0];
    const float* memory = (const float*)d_in[1];
    const float* cba    = (const float*)d_in[2];
    const float* cbb    = (const float*)d_in[3];
    const float* Wa     = (const float*)d_in[4];
    const float* ba     = (const float*)d_in[5];
    const float* Wb     = (const float*)d_in[6];
    const float* bb     = (const float*)d_in[7];
    const float* Wv     = (const float*)d_in[8];
    const float* bv     = (const float*)d_in[9];
    const float* Wo     = (const float*)d_in[10];
    const float* bo     = (const float*)d_in[11];
    const float* Wg     = (const float*)d_in[12];
    const float* bg     = (const float*)d_in[13];

    float* ws   = (float*)d_ws;
    float* outX = (float*)d_out;                       // [4,2048,1024]
    float* outM = outX + (size_t)BB * SS * DD;         // [4,65536,256]

    k_xsummary<<<(MP * DD) / 256, 256, 0, stream>>>(x, ws);
    k_gate<<<BB, 256, 0, stream>>>(Wg, bg, ws);
    k_wmma_qv<<<32, 32, 0, stream>>>(Wa, ba, Wb, bb, Wv, bv, ws);
    k_wmma_sim<<<32, 32, 0, stream>>>(cba, cbb, ws);
    k_topk<<<BB * 2, 256, 0, stream>>>(ws);
    k_softmax<<<BB, 1024, 0, stream>>>(ws);
    k_readout_part<<<BB * 8, SDIM, 0, stream>>>(memory, ws);
    k_readout_red<<<MP, SDIM, 0, stream>>>(ws);
    k_wmma_rp<<<64, 32, 0, stream>>>(Wo, bo, ws);
    k_xaug<<<(BB * SS * DD / 4) / 256, 256, 0, stream>>>(x, ws, outX);
    k_memupd<<<((size_t)BB * MSLOTS * SDIM / 4) / 256, 256, 0, stream>>>(memory, ws, outM);
}